// SegmenterTorch_56092272886447
// MI455X (gfx1250) — compile-verified
//
#include <hip/hip_runtime.h>
#include <cstdint>
#include <cstddef>

#define FRAME 1024
#define HOP 512
#define SEGS_PER_WG 16
#define THREADS 256

typedef float __attribute__((ext_vector_type(4))) f32x4;
typedef int   __attribute__((ext_vector_type(4))) v4i;

#if __has_builtin(__builtin_amdgcn_global_load_async_to_lds_b128)
#define HAS_ASYNC_LDS 1
#else
#define HAS_ASYNC_LDS 0
#endif

typedef __attribute__((address_space(1))) v4i* gv4ip;
typedef __attribute__((address_space(3))) v4i* lv4ip;

__device__ __forceinline__ void async_copy_b128(const float* gsrc, float* ldst) {
#if HAS_ASYNC_LDS
    // (global v4i*, lds v4i*, imm offset, imm cpol)
    __builtin_amdgcn_global_load_async_to_lds_b128(
        (gv4ip)(v4i*)gsrc, (lv4ip)(v4i*)ldst, /*offset=*/0, /*cpol=*/0);
#else
    *(f32x4*)ldst = *(const f32x4*)gsrc;
#endif
}

__device__ __forceinline__ void wait_async_done() {
#if HAS_ASYNC_LDS
#if __has_builtin(__builtin_amdgcn_s_wait_asynccnt)
    __builtin_amdgcn_s_wait_asynccnt(0);
#else
    asm volatile("s_wait_asynccnt 0x0" ::: "memory");
#endif
#endif
}

// One workgroup: one batch row x one block of SEGS_PER_WG consecutive segments.
// The input tile (SEGS_PER_WG+1)*HOP samples = block + 1-hop halo is staged to
// LDS once via async DMA, giving each HBM byte exactly one read despite the
// 2x frame overlap. Window float4s are loop-invariant per thread -> registers.
// Output is written once with non-temporal b128 stores.
__global__ __launch_bounds__(THREADS) void segwin_kernel(
    const float* __restrict__ x,
    const float* __restrict__ window,
    const float* __restrict__ prewindow,
    const float* __restrict__ postwindow,
    float* __restrict__ out,
    int n_seg, int samples)
{
    __shared__ float lds_x[(SEGS_PER_WG + 1) * HOP];  // 34 KB

    const int tid  = threadIdx.x;
    const int b    = blockIdx.y;
    const int seg0 = blockIdx.x * SEGS_PER_WG;
    const int nseg_blk = min(SEGS_PER_WG, n_seg - seg0);
    if (nseg_blk <= 0) return;

    // ---- async stage: input block + 1-hop halo ----
    const float* xrow = x + (size_t)b * (size_t)samples + (size_t)seg0 * HOP;
    const int nvec = (nseg_blk + 1) * (HOP / 4);
    for (int j = tid; j < nvec; j += THREADS) {
        async_copy_b128(xrow + (size_t)j * 4, lds_x + j * 4);
    }

    // ---- window float4 for this thread's fixed frame positions (registers) ----
    const int c4 = tid * 4;
    const f32x4 wv_mid = *(const f32x4*)(window + c4);
    f32x4 wv_pre  = wv_mid;
    f32x4 wv_post = wv_mid;
    if (seg0 == 0)                  wv_pre  = *(const f32x4*)(prewindow + c4);
    if (seg0 + nseg_blk == n_seg)   wv_post = *(const f32x4*)(postwindow + c4);

    wait_async_done();   // each wave drains its own ASYNCcnt
    __syncthreads();     // then all waves see the full LDS tile

    // ---- compute + stream out: one ds_load_b128 + mul + NT b128 store / seg ----
    float* orow = out + ((size_t)b * (size_t)n_seg + (size_t)seg0) * FRAME;
    #pragma unroll 4
    for (int s = 0; s < nseg_blk; ++s) {
        const int g = seg0 + s;
        f32x4 wv = (g == 0) ? wv_pre : ((g == n_seg - 1) ? wv_post : wv_mid);
        f32x4 xv = *(const f32x4*)(lds_x + s * HOP + c4);
        f32x4 r  = xv * wv;
        __builtin_nontemporal_store(r, (f32x4*)(orow + (size_t)s * FRAME + c4));
    }
}

extern "C" void kernel_launch(void* const* d_in, const int* in_sizes, int n_in,
                              void* d_out, int out_size, void* d_ws, size_t ws_size,
                              hipStream_t stream) {
    const float* x    = (const float*)d_in[0];
    const float* w    = (const float*)d_in[1];
    const float* pre  = (const float*)d_in[2];
    const float* post = (const float*)d_in[3];
    // d_in[4] (hop) / d_in[5] (frame) are device scalars; recover shapes on host:
    const long long in0   = (long long)in_sizes[0];      // BATCH * SAMPLES
    const long long frame = (long long)in_sizes[1];      // window length = FRAME
    const long long hop   = frame / 2;                   // per reference: HOP = FRAME/2
    // out = B*n_seg*F, n_seg = S/h - 1  =>  out = 2*in0 - B*F
    long long batch = (2 * in0 - (long long)out_size) / frame;
    if (batch <= 0) batch = 1;
    const long long samples = in0 / batch;
    const int n_seg = (int)(samples / hop) - (int)(frame / hop) + 1;

    float* out = (float*)d_out;
    const int seg_blocks = (n_seg + SEGS_PER_WG - 1) / SEGS_PER_WG;
    dim3 grid(seg_blocks, (unsigned)batch, 1);
    dim3 block(THREADS, 1, 1);
    segwin_kernel<<<grid, block, 0, stream>>>(x, w, pre, post, out,
                                              n_seg, (int)samples);
}